// AdaptToLine_40956808135194
// MI455X (gfx1250) — compile-verified
//
#include <hip/hip_runtime.h>
#include <cstdint>

typedef float v2f __attribute__((ext_vector_type(2)));
typedef float v4f __attribute__((ext_vector_type(4)));
typedef float v8f __attribute__((ext_vector_type(8)));

// Global (addrspace(1)) pointer types: force global_load/global_store lowering
// even when the base pointer is produced by a select/phi (which blocks the
// compiler's flat->global inference and otherwise yields flat_* ops that tie
// up DScnt and the LDS path).
typedef __attribute__((address_space(1))) const float*    gcf_p;
typedef __attribute__((address_space(1))) float*          gf_p;
typedef __attribute__((address_space(1))) const v4f*      gcv4_p;
typedef __attribute__((address_space(1))) v4f*            gv4_p;
typedef __attribute__((address_space(1))) const long long* gci64_p;

static __device__ __forceinline__ gcf_p  gc(const float* p) { return (gcf_p)(unsigned long long)p; }
static __device__ __forceinline__ gf_p   gm(float* p)       { return (gf_p)(unsigned long long)p; }
static __device__ __forceinline__ gci64_p gi(const long long* p) { return (gci64_p)(unsigned long long)p; }

#define N_NODES 100000
#define E_EDGES 1600000
#define IN_C    128
#define OUT_C   64
#define EDGE_D  64

// ---------------------------------------------------------------------------
// Kernel A: node projections  k,q,v,skip = x @ W_* + b_*
// One wave computes a 16-node x 64-out tile for all four weight matrices.
// A-tile (16x128) preloaded once into registers (v2f a[32] = 64 VGPRs); the
// four weights processed sequentially with 4 live v8f accumulators each.
//
// WMMA f32 16x16x4 layouts:
//  A (16x4): lanes0-15 row=m, VGPR0=K+0 VGPR1=K+1; lanes16-31 K+2/K+3
//  B (4x16): VGPR0: lanes0-15 K+0 / lanes16-31 K+2; VGPR1: K+1 / K+3
//  C (16x16): VGPR r: lanes0-15 M=r, lanes16-31 M=r+8; N = lane%16
// ---------------------------------------------------------------------------
__global__ void node_proj_kernel(const float* __restrict__ x,
    const float* __restrict__ Wk, const float* __restrict__ bk,
    const float* __restrict__ Wq, const float* __restrict__ bq,
    const float* __restrict__ Wv, const float* __restrict__ bv,
    const float* __restrict__ Wsk, const float* __restrict__ bsk,
    float* __restrict__ kOut, float* __restrict__ qOut,
    float* __restrict__ vOut, float* __restrict__ sOut)
{
    const int lane = threadIdx.x & 31;
    const int wave = threadIdx.x >> 5;
    const int tile = blockIdx.x * (blockDim.x >> 5) + wave;
    const int m0   = tile * 16;
    if (m0 >= N_NODES) return;            // wave-uniform: EXEC stays all-1s
    const int hi   = lane >> 4;           // 0 = lanes 0-15, 1 = lanes 16-31
    const int lsub = lane & 15;

    // Preload the whole A tile for K=0..127 (read x from HBM exactly once)
    v2f a[IN_C / 4];
    {
        gcf_p arow = gc(x) + (size_t)(m0 + lsub) * IN_C + hi * 2;
#pragma unroll
        for (int kk = 0; kk < IN_C / 4; ++kk) {
            a[kk][0] = arow[kk * 4 + 0];
            a[kk][1] = arow[kk * 4 + 1];
        }
    }

    gcf_p Ws[4] = {gc(Wk), gc(Wq), gc(Wv), gc(Wsk)};
    gcf_p bs[4] = {gc(bk), gc(bq), gc(bv), gc(bsk)};
    gf_p outs[4] = {gm(kOut), gm(qOut), gm(vOut), gm(sOut)};

#pragma unroll 1
    for (int w = 0; w < 4; ++w) {
        gcf_p W = Ws[w];
        v8f acc[4];
        {
            gcf_p bb = bs[w];
#pragma unroll
            for (int j = 0; j < 4; ++j) {
                const float b = bb[j * 16 + lsub];
#pragma unroll
                for (int r = 0; r < 8; ++r) acc[j][r] = b;
            }
        }

#pragma unroll 4
        for (int kk = 0; kk < IN_C / 4; ++kk) {
            const int k = kk * 4;
            gcf_p Wr0 = W + (size_t)(k + hi * 2 + 0) * OUT_C + lsub;
            gcf_p Wr1 = W + (size_t)(k + hi * 2 + 1) * OUT_C + lsub;
            v2f b[4];
#pragma unroll
            for (int j = 0; j < 4; ++j) {        // batch all 4 B loads first
                b[j][0] = Wr0[j * 16];
                b[j][1] = Wr1[j * 16];
            }
#pragma unroll
            for (int j = 0; j < 4; ++j)
                acc[j] = __builtin_amdgcn_wmma_f32_16x16x4_f32(
                    false, a[kk], false, b[j], (short)0, acc[j], false, false);
        }

        gf_p o = outs[w];
#pragma unroll
        for (int j = 0; j < 4; ++j)
#pragma unroll
            for (int r = 0; r < 8; ++r) {
                const int m = m0 + r + hi * 8;
                o[(size_t)m * OUT_C + j * 16 + lsub] = acc[j][r];
            }
    }
}

// ---------------------------------------------------------------------------
// Kernel B: fused edge pipeline. One wave handles 16 edges:
//   e = edge_attr_tile @ W_edge + b_edge        (WMMA f32 16x16x4, K=64)
//   gate = sigmoid(k[dst] + q[src] + e)
//   atomicAdd(agg[dst], gate * v[src])          (L2-resident fp32 atomics)
//   + write edge_attr passthrough copy to d_out (rows already in cache)
// ---------------------------------------------------------------------------
__global__ void edge_kernel(const float* __restrict__ edge_attr,
    const long long* __restrict__ eidx,          // [2][E], row0=src, row1=dst
    const float* __restrict__ We, const float* __restrict__ be,
    const float* __restrict__ kArr, const float* __restrict__ qArr,
    const float* __restrict__ vArr,
    float* __restrict__ agg, float* __restrict__ eattr_out)
{
    const int lane = threadIdx.x;
    const int hi   = lane >> 4;
    const int lsub = lane & 15;
    const int e0   = blockIdx.x * 16;

    gcf_p ea = gc(edge_attr);
    gcf_p Wg = gc(We);

    v8f acc[4];
    {
        gcf_p bb = gc(be);
#pragma unroll
        for (int j = 0; j < 4; ++j) {
            const float b = bb[j * 16 + lsub];
#pragma unroll
            for (int r = 0; r < 8; ++r) acc[j][r] = b;
        }
    }

#pragma unroll 4
    for (int k = 0; k < EDGE_D; k += 4) {
        gcf_p arow = ea + (size_t)(e0 + lsub) * EDGE_D + k + hi * 2;
        v2f a; a[0] = arow[0]; a[1] = arow[1];
        gcf_p Wr0 = Wg + (size_t)(k + hi * 2 + 0) * OUT_C + lsub;
        gcf_p Wr1 = Wg + (size_t)(k + hi * 2 + 1) * OUT_C + lsub;
        v2f b[4];
#pragma unroll
        for (int j = 0; j < 4; ++j) {            // batch B loads before WMMAs
            b[j][0] = Wr0[j * 16];
            b[j][1] = Wr1[j * 16];
        }
#pragma unroll
        for (int j = 0; j < 4; ++j)
            acc[j] = __builtin_amdgcn_wmma_f32_16x16x4_f32(
                false, a, false, b[j], (short)0, acc[j], false, false);
    }

    // Passthrough copy of edge_attr tile (16x64 fp32 = 8 x b128 per lane)
    {
        gcv4_p srcp = (gcv4_p)(ea + (size_t)e0 * EDGE_D);
        gv4_p  dstp = (gv4_p)(gm(eattr_out) + (size_t)e0 * EDGE_D);
#pragma unroll
        for (int t = 0; t < 8; ++t) dstp[lane + t * 32] = srcp[lane + t * 32];
    }

    // Each lane owns edges e0 + r + hi*8, r = 0..7 (matches C-matrix striping)
    gci64_p srcIdx = gi(eidx);
    gci64_p dstIdx = gi(eidx) + (size_t)E_EDGES;
    int se[8], de[8];
#pragma unroll
    for (int r = 0; r < 8; ++r) {
        const int e = e0 + r + hi * 8;
        se[r] = (int)srcIdx[e];
        de[r] = (int)dstIdx[e];
    }

    gcf_p kg = gc(kArr);
    gcf_p qg = gc(qArr);
    gcf_p vg = gc(vArr);

#pragma unroll
    for (int j = 0; j < 4; ++j) {
        const int ch = j * 16 + lsub;
#pragma unroll
        for (int r = 0; r < 8; ++r) {
            const float kq = kg[(size_t)de[r] * OUT_C + ch] +
                             qg[(size_t)se[r] * OUT_C + ch];
            const float g   = 1.0f / (1.0f + __expf(-(kq + acc[j][r])));
            const float msg = g * vg[(size_t)se[r] * OUT_C + ch];
            __hip_atomic_fetch_add(&agg[(size_t)de[r] * OUT_C + ch], msg,
                                   __ATOMIC_RELAXED, __HIP_MEMORY_SCOPE_AGENT);
        }
    }
}

// ---------------------------------------------------------------------------
// Kernel C: out = relu(agg + skip) (b128-vectorized); copy u (64 floats)
// ---------------------------------------------------------------------------
__global__ void finalize_kernel(const float* __restrict__ aggf,
                                const float* __restrict__ skipf,
                                const float* __restrict__ u,
                                float* __restrict__ outf,
                                float* __restrict__ uOut)
{
    const int i = blockIdx.x * blockDim.x + threadIdx.x;
    gcv4_p agg  = (gcv4_p)gc(aggf);
    gcv4_p skip = (gcv4_p)gc(skipf);
    gv4_p  out  = (gv4_p)gm(outf);
    if (i < (N_NODES * OUT_C) / 4) {
        const v4f a = agg[i];
        const v4f s = skip[i];
        v4f r;
#pragma unroll
        for (int c = 0; c < 4; ++c) r[c] = fmaxf(a[c] + s[c], 0.0f);
        out[i] = r;
    }
    if (blockIdx.x == 0 && threadIdx.x < 64) gm(uOut)[threadIdx.x] = gc(u)[threadIdx.x];
}

// ---------------------------------------------------------------------------
extern "C" void kernel_launch(void* const* d_in, const int* in_sizes, int n_in,
                              void* d_out, int out_size, void* d_ws, size_t ws_size,
                              hipStream_t stream)
{
    const float*     x          = (const float*)d_in[0];
    const long long* edge_index = (const long long*)d_in[1];
    const float*     edge_attr  = (const float*)d_in[2];
    const float*     u          = (const float*)d_in[3];
    const float* Wk  = (const float*)d_in[4];  const float* bk  = (const float*)d_in[5];
    const float* Wq  = (const float*)d_in[6];  const float* bq  = (const float*)d_in[7];
    const float* Wv  = (const float*)d_in[8];  const float* bv  = (const float*)d_in[9];
    const float* We  = (const float*)d_in[10]; const float* be  = (const float*)d_in[11];
    const float* Wsk = (const float*)d_in[12]; const float* bsk = (const float*)d_in[13];

    // Workspace: k | q | v | skip | agg   (5 x 25.6 MB fp32)
    float* kArr = (float*)d_ws;
    float* qArr = kArr + (size_t)N_NODES * OUT_C;
    float* vArr = qArr + (size_t)N_NODES * OUT_C;
    float* sArr = vArr + (size_t)N_NODES * OUT_C;
    float* agg  = sArr + (size_t)N_NODES * OUT_C;

    // Output layout: out [N*64] | edge_attr copy [E*64] | u [64]
    float* out       = (float*)d_out;
    float* eattr_out = out + (size_t)N_NODES * OUT_C;
    float* uOut      = eattr_out + (size_t)E_EDGES * OUT_C;

    (void)hipMemsetAsync(agg, 0, (size_t)N_NODES * OUT_C * sizeof(float), stream);

    // A: 6250 16-node tiles, 4 waves (128 threads) per block
    {
        const int tiles = (N_NODES + 15) / 16;          // 6250
        const int wavesPerBlock = 4;
        const int blocks = (tiles + wavesPerBlock - 1) / wavesPerBlock;
        node_proj_kernel<<<blocks, 32 * wavesPerBlock, 0, stream>>>(
            x, Wk, bk, Wq, bq, Wv, bv, Wsk, bsk, kArr, qArr, vArr, sArr);
    }

    // B: 100000 16-edge tiles, one wave each
    edge_kernel<<<E_EDGES / 16, 32, 0, stream>>>(
        edge_attr, edge_index, We, be, kArr, qArr, vArr, agg, eattr_out);

    // C: elementwise finalize (b128)
    {
        const int total4 = (N_NODES * OUT_C) / 4;
        finalize_kernel<<<(total4 + 255) / 256, 256, 0, stream>>>(
            agg, sArr, u, out, uOut);
    }
}